// GCN_72421738545283
// MI455X (gfx1250) — compile-verified
//
#include <hip/hip_runtime.h>

typedef __attribute__((ext_vector_type(2))) float v2f;
typedef __attribute__((ext_vector_type(8))) float v8f;

#define DHID 64
#define DOUT 16

// ---------------------------------------------------------------------------
// Degree / normalization kernels
// ---------------------------------------------------------------------------
__global__ void deg_init_kernel(float* __restrict__ deg, int n) {
    int i = blockIdx.x * blockDim.x + threadIdx.x;
    if (i < n) deg[i] = 1.0f;  // self-loop contributes 1 to every node's degree
}

__global__ void deg_count_kernel(const long long* __restrict__ dst,
                                 float* __restrict__ deg, int nedges) {
    int e = blockIdx.x * blockDim.x + threadIdx.x;
    if (e < nedges) atomicAdd(&deg[(int)dst[e]], 1.0f);
}

__global__ void dinv_kernel(const float* __restrict__ deg,
                            float* __restrict__ dinv, int n) {
    int i = blockIdx.x * blockDim.x + threadIdx.x;
    if (i < n) dinv[i] = rsqrtf(deg[i]);  // deg >= 1 always
}

// ---------------------------------------------------------------------------
// WMMA GEMM: Y[n x 64] = X[n x 64] * W[64 x 64]
// One wave computes a 16-row x 64-col strip (4 x 16x16 tiles), K stepped by 4
// using V_WMMA_F32_16X16X4_F32 (full fp32 precision).
// A-frag layout (ISA 7.12.2, 32-bit A 16x4): lanes 0-15 = rows, lane>>4
// selects K pair {0,1} vs {2,3}; VGPR0=Klo, VGPR1=Khi.
// B/C layout: row striped across lanes within a VGPR.
// ---------------------------------------------------------------------------
__global__ void __launch_bounds__(128) gemm64x64_kernel(
    const float* __restrict__ X, const float* __restrict__ W,
    float* __restrict__ Y, int nrows) {
    const int wave = threadIdx.x >> 5;          // wave32: 4 waves / block
    const int lane = threadIdx.x & 31;
    const int row0 = (blockIdx.x * 4 + wave) * 16;
    if (row0 >= nrows) return;                   // wave-uniform: EXEC stays ~0

    const int m = lane & 15;
    const int khalf = lane >> 4;                 // 0 or 1

    int arow = row0 + m;
    if (arow >= nrows) arow = nrows - 1;         // clamp reads; stores predicated
    const float* xrow = X + (size_t)arow * DHID;

    v8f acc0 = {}, acc1 = {}, acc2 = {}, acc3 = {};

    for (int k0 = 0; k0 < DHID; k0 += 4) {
        v2f a;
        a.x = xrow[k0 + 2 * khalf + 0];
        a.y = xrow[k0 + 2 * khalf + 1];
        const float* w0 = W + (size_t)(k0 + 2 * khalf + 0) * DHID;
        const float* w1 = W + (size_t)(k0 + 2 * khalf + 1) * DHID;
        v2f b0, b1, b2, b3;
        b0.x = w0[m +  0]; b0.y = w1[m +  0];
        b1.x = w0[m + 16]; b1.y = w1[m + 16];
        b2.x = w0[m + 32]; b2.y = w1[m + 32];
        b3.x = w0[m + 48]; b3.y = w1[m + 48];
        acc0 = __builtin_amdgcn_wmma_f32_16x16x4_f32(false, a, false, b0, (short)0, acc0, false, false);
        acc1 = __builtin_amdgcn_wmma_f32_16x16x4_f32(false, a, false, b1, (short)0, acc1, false, false);
        acc2 = __builtin_amdgcn_wmma_f32_16x16x4_f32(false, a, false, b2, (short)0, acc2, false, false);
        acc3 = __builtin_amdgcn_wmma_f32_16x16x4_f32(false, a, false, b3, (short)0, acc3, false, false);
    }

#pragma unroll
    for (int v = 0; v < 8; ++v) {
        int r = row0 + v + 8 * khalf;            // C layout: VGPR v -> row v / v+8
        if (r < nrows) {
            float* yr = Y + (size_t)r * DHID;
            yr[m +  0] = acc0[v];
            yr[m + 16] = acc1[v];
            yr[m + 32] = acc2[v];
            yr[m + 48] = acc3[v];
        }
    }
}

// ---------------------------------------------------------------------------
// WMMA GEMM classifier: Y[n x 16] = X[n x 64] * Wc[64 x 16] + bc
// ---------------------------------------------------------------------------
__global__ void __launch_bounds__(128) gemm64x16_kernel(
    const float* __restrict__ X, const float* __restrict__ Wc,
    const float* __restrict__ bc, float* __restrict__ Y, int nrows) {
    const int wave = threadIdx.x >> 5;
    const int lane = threadIdx.x & 31;
    const int row0 = (blockIdx.x * 4 + wave) * 16;
    if (row0 >= nrows) return;

    const int m = lane & 15;
    const int khalf = lane >> 4;

    int arow = row0 + m;
    if (arow >= nrows) arow = nrows - 1;
    const float* xrow = X + (size_t)arow * DHID;

    v8f acc = {};
    for (int k0 = 0; k0 < DHID; k0 += 4) {
        v2f a;
        a.x = xrow[k0 + 2 * khalf + 0];
        a.y = xrow[k0 + 2 * khalf + 1];
        v2f b;
        b.x = Wc[(size_t)(k0 + 2 * khalf + 0) * DOUT + m];
        b.y = Wc[(size_t)(k0 + 2 * khalf + 1) * DOUT + m];
        acc = __builtin_amdgcn_wmma_f32_16x16x4_f32(false, a, false, b, (short)0, acc, false, false);
    }

    const float bias = bc[m];
#pragma unroll
    for (int v = 0; v < 8; ++v) {
        int r = row0 + v + 8 * khalf;
        if (r < nrows) Y[(size_t)r * DOUT + m] = acc[v] + bias;
    }
}

// ---------------------------------------------------------------------------
// Self-loop init: agg[i,:] = h[i,:] * dinv[i]^2   (also zero-fills/initializes)
// ---------------------------------------------------------------------------
__global__ void selfloop_init_kernel(const float* __restrict__ h,
                                     const float* __restrict__ dinv,
                                     float* __restrict__ agg, int total) {
    int i = blockIdx.x * blockDim.x + threadIdx.x;
    if (i < total) {
        float di = dinv[i >> 6];  // 64 feats per node
        agg[i] = h[i] * di * di;
    }
}

// ---------------------------------------------------------------------------
// Edge propagation: 16 threads per edge, float4 per thread.
// agg[dst] += h[src] * dinv[src]*dinv[dst]  via global_atomic_add_f32.
// Both agg buffers (25.6 MB each) are L2-resident on MI455X (192 MB L2),
// so these atomics resolve at L2 bandwidth.
// ---------------------------------------------------------------------------
__global__ void edge_prop_kernel(const float* __restrict__ h,
                                 const long long* __restrict__ src,
                                 const long long* __restrict__ dst,
                                 const float* __restrict__ dinv,
                                 float* __restrict__ agg, int nedges) {
    int t = blockIdx.x * blockDim.x + threadIdx.x;
    int e = t >> 4;
    if (e >= nedges) return;
    int f = (t & 15) * 4;
    int s = (int)src[e];
    int d = (int)dst[e];
    float nrm = dinv[s] * dinv[d];
    float4 v = *(const float4*)(h + (size_t)s * DHID + f);
    float* out = agg + (size_t)d * DHID + f;
    atomicAdd(out + 0, v.x * nrm);
    atomicAdd(out + 1, v.y * nrm);
    atomicAdd(out + 2, v.z * nrm);
    atomicAdd(out + 3, v.w * nrm);
}

// ---------------------------------------------------------------------------
// Bias + ReLU elementwise
// ---------------------------------------------------------------------------
__global__ void bias_relu_kernel(const float* __restrict__ agg,
                                 const float* __restrict__ bias,
                                 float* __restrict__ out, int total) {
    int i = blockIdx.x * blockDim.x + threadIdx.x;
    if (i < total) {
        float v = agg[i] + bias[i & 63];
        out[i] = v > 0.0f ? v : 0.0f;
    }
}

// ---------------------------------------------------------------------------
// Orchestration
// ---------------------------------------------------------------------------
extern "C" void kernel_launch(void* const* d_in, const int* in_sizes, int n_in,
                              void* d_out, int out_size, void* d_ws, size_t ws_size,
                              hipStream_t stream) {
    const float*     x   = (const float*)d_in[0];
    const long long* ei  = (const long long*)d_in[1];   // int64 [2, E]
    const float*     W1  = (const float*)d_in[2];
    const float*     b1  = (const float*)d_in[3];
    const float*     W2  = (const float*)d_in[4];
    const float*     b2  = (const float*)d_in[5];
    const float*     Wc  = (const float*)d_in[6];
    const float*     bc  = (const float*)d_in[7];
    float*           out = (float*)d_out;

    const int N = in_sizes[0] / DHID;
    const int E = in_sizes[1] / 2;
    const long long* src = ei;
    const long long* dst = ei + E;

    // Workspace layout (floats): deg[N] | dinv[N] | bufA[N*64] | bufB[N*64]
    float* deg  = (float*)d_ws;
    float* dinv = deg + N;
    float* bufA = dinv + N;
    float* bufB = bufA + (size_t)N * DHID;

    const int T = 256;
    const int nFeat = N * DHID;
    dim3 blk(T);
    dim3 gN((N + T - 1) / T);
    dim3 gE((E + T - 1) / T);
    dim3 gF((nFeat + T - 1) / T);
    dim3 gEdge(((size_t)E * 16 + T - 1) / T);
    dim3 gGemm((N / 16 + 3) / 4);   // 4 waves (16-row strips) per 128-thread block

    // Degree + D^-1/2
    deg_init_kernel<<<gN, blk, 0, stream>>>(deg, N);
    deg_count_kernel<<<gE, blk, 0, stream>>>(dst, deg, E);
    dinv_kernel<<<gN, blk, 0, stream>>>(deg, dinv, N);

    // Layer 1: h1 = x@W1 -> bufA ; agg -> bufB ; relu(agg+b1) -> bufA
    gemm64x64_kernel<<<gGemm, 128, 0, stream>>>(x, W1, bufA, N);
    selfloop_init_kernel<<<gF, blk, 0, stream>>>(bufA, dinv, bufB, nFeat);
    edge_prop_kernel<<<gEdge, blk, 0, stream>>>(bufA, src, dst, dinv, bufB, E);
    bias_relu_kernel<<<gF, blk, 0, stream>>>(bufB, b1, bufA, nFeat);

    // Layer 2: h2 = h1r@W2 -> bufB ; agg -> bufA ; relu(agg+b2) -> bufB
    gemm64x64_kernel<<<gGemm, 128, 0, stream>>>(bufA, W2, bufB, N);
    selfloop_init_kernel<<<gF, blk, 0, stream>>>(bufB, dinv, bufA, nFeat);
    edge_prop_kernel<<<gEdge, blk, 0, stream>>>(bufB, src, dst, dinv, bufA, E);
    bias_relu_kernel<<<gF, blk, 0, stream>>>(bufA, b2, bufB, nFeat);

    // Classifier: out = h2r@Wc + bc
    gemm64x16_kernel<<<gGemm, 128, 0, stream>>>(bufB, Wc, bc, out, N);
}